// ToxGNN_42210938585221
// MI455X (gfx1250) — compile-verified
//
#include <hip/hip_runtime.h>
#include <hip/hip_bf16.h>

typedef __attribute__((ext_vector_type(2))) float    v2f;
typedef __attribute__((ext_vector_type(8))) float    v8f;
typedef __attribute__((ext_vector_type(4))) unsigned v4u;
typedef __attribute__((ext_vector_type(8))) int      v8i;
typedef __attribute__((ext_vector_type(4))) int      v4i;

#define HID 64

// ---------------- utility kernels ----------------

__global__ void zero_kernel(float* __restrict__ p, long long n) {
    long long i = (long long)blockIdx.x * blockDim.x + threadIdx.x;
    long long stride = (long long)gridDim.x * blockDim.x;
    for (; i < n; i += stride) p[i] = 0.0f;
}

__global__ void deg_kernel(const int* __restrict__ row, float* __restrict__ deg, int E) {
    int e = blockIdx.x * blockDim.x + threadIdx.x;
    if (e < E) atomicAdd(&deg[row[e]], 1.0f);
}

__global__ void dinv_kernel(float* __restrict__ d, int N) {
    int i = blockIdx.x * blockDim.x + threadIdx.x;
    if (i < N) d[i] = rsqrtf(d[i] + 1.0f);
}

// ---------------- WMMA GEMM: C[M,64] = A[M,K] @ B[K,64] ----------------
// wave32; one wave computes a 16-row M tile x all 64 output columns.
// B panel (<= 208*64 floats = 53 KB) is staged in LDS (320 KB/WGP) — via the
// Tensor Data Mover when the builtin is available (one wave issues a 1D-tile
// descriptor per CDNA5 ISA, then s_wait_tensorcnt 0).
__global__ void gemm64_wmma(const float* __restrict__ A,
                            const float* __restrict__ B,
                            float* __restrict__ C, int M, int K) {
    extern __shared__ float lds[];
    const int nElem = K * HID;

#if __has_builtin(__builtin_amdgcn_tensor_load_to_lds) && \
    __has_builtin(__builtin_amdgcn_s_wait_tensorcnt)
    if (threadIdx.x < 32) {
        // D# group 0: count=1 | lds_addr | global_addr[56:0] | type=2
        unsigned lds_off = (unsigned)(unsigned long long)(size_t)lds;
        unsigned long long ga = (unsigned long long)(const void*)B;
        v4u g0;
        g0.x = 1u;                                   // count = 1 valid descriptor
        g0.y = lds_off;                              // lds_addr (bytes)
        g0.z = (unsigned)(ga & 0xFFFFFFFFull);       // global_addr[31:0]
        g0.w = (unsigned)((ga >> 32) & 0x01FFFFFFu)  // global_addr[56:32]
               | (2u << 30);                         // type = 2 ("image")
        // D# group 1: 1-D tile of nElem 4-byte elements
        v8i g1;
        g1[0] = 0x00020000;                          // wg_mask=0, data_size=2 (4B)
        g1[1] = (int)((nElem & 0xFFFF) << 16);       // tensor_dim0[15:0]
        g1[2] = (int)((unsigned)(nElem >> 16) | (1u << 16)); // dim0[31:16] | tensor_dim1=1
        g1[3] = (int)((nElem & 0xFFFF) << 16);       // tile_dim0 = nElem
        g1[4] = 1;                                   // tile_dim1 = 1, tile_dim2 = 0
        g1[5] = nElem;                               // tensor_dim0_stride[31:0]
        g1[6] = 0;                                   // stride hi / dim1_stride lo
        g1[7] = 0;
        v4i gz = {0, 0, 0, 0};
#if defined(__clang_major__) && __clang_major__ >= 23
        v8i gz8 = {0, 0, 0, 0, 0, 0, 0, 0};
        __builtin_amdgcn_tensor_load_to_lds(g0, g1, gz, gz, gz8, 0);
#else
        __builtin_amdgcn_tensor_load_to_lds(g0, g1, gz, gz, 0);
#endif
        __builtin_amdgcn_s_wait_tensorcnt(0);
    }
    __syncthreads();
#else
    for (int i = threadIdx.x; i < nElem; i += blockDim.x) lds[i] = B[i];
    __syncthreads();
#endif

    const int wave  = threadIdx.x >> 5;
    const int lane  = threadIdx.x & 31;
    const int mtile = blockIdx.x * (blockDim.x >> 5) + wave;
    if (mtile * 16 >= M) return;

    const int mrow = mtile * 16 + (lane & 15);
    const int koff = (lane >> 4) << 1;            // 0 for lanes 0-15, 2 for lanes 16-31
    const int ncol = lane & 15;
    const float* __restrict__ arow = A + (long long)mrow * K;

    v8f acc[4] = {v8f{}, v8f{}, v8f{}, v8f{}};

    for (int k0 = 0; k0 < K; k0 += 4) {
        v2f a;
        a.x = arow[k0 + koff];
        a.y = arow[k0 + koff + 1];
        const float* __restrict__ brow0 = &lds[(k0 + koff) * HID + ncol];
        const float* __restrict__ brow1 = brow0 + HID;
#pragma unroll
        for (int nt = 0; nt < 4; ++nt) {
            v2f b;
            b.x = brow0[nt * 16];
            b.y = brow1[nt * 16];
            acc[nt] = __builtin_amdgcn_wmma_f32_16x16x4_f32(
                false, a, false, b, (short)0, acc[nt], false, false);
        }
    }

    const int rbase = mtile * 16 + ((lane >> 4) << 3);
#pragma unroll
    for (int nt = 0; nt < 4; ++nt) {
#pragma unroll
        for (int r = 0; r < 8; ++r) {
            C[(long long)(rbase + r) * HID + nt * 16 + ncol] = acc[nt][r];
        }
    }
}

// agg[i][f0..f0+3] = dinv[i]^2 * h[i][..] + b[..]   (self-loop term + bias), float4
__global__ void self_bias_kernel(float4* __restrict__ agg, const float4* __restrict__ h,
                                 const float* __restrict__ dinv, const float4* __restrict__ b,
                                 long long total /* N*16 */) {
    long long t = (long long)blockIdx.x * blockDim.x + threadIdx.x;
    if (t < total) {
        long long i = t >> 4;
        int q = (int)(t & 15);
        float d2 = dinv[i] * dinv[i];
        float4 hv = h[t];
        float4 bv = b[q];
        agg[t] = make_float4(d2 * hv.x + bv.x, d2 * hv.y + bv.y,
                             d2 * hv.z + bv.z, d2 * hv.w + bv.w);
    }
}

// agg[row[e]][4q..4q+3] += dinv[row]*dinv[col] * h[col][4q..4q+3]
__global__ void scatter_kernel(const int* __restrict__ row, const int* __restrict__ col,
                               const float* __restrict__ dinv, const float* __restrict__ h,
                               float* __restrict__ agg, long long total /* E*16 */) {
    long long t = (long long)blockIdx.x * blockDim.x + threadIdx.x;
    if (t < total) {
        long long e = t >> 4;
        int q = (int)(t & 15);
        int r = row[e];
        int c = col[e];
        float nrm = dinv[r] * dinv[c];
        float4 hv = ((const float4*)(h + (long long)c * HID))[q];
        float* dst = agg + (long long)r * HID + q * 4;
        atomicAdd(dst + 0, nrm * hv.x);
        atomicAdd(dst + 1, nrm * hv.y);
        atomicAdd(dst + 2, nrm * hv.z);
        atomicAdd(dst + 3, nrm * hv.w);
    }
}

__global__ void relu_kernel(float4* __restrict__ p, long long n4) {
    long long t = (long long)blockIdx.x * blockDim.x + threadIdx.x;
    if (t < n4) {
        float4 v = p[t];
        p[t] = make_float4(fmaxf(v.x, 0.f), fmaxf(v.y, 0.f),
                           fmaxf(v.z, 0.f), fmaxf(v.w, 0.f));
    }
}

__global__ void bias_relu_kernel(float4* __restrict__ p, const float4* __restrict__ b,
                                 long long n4 /* rows*16 */) {
    long long t = (long long)blockIdx.x * blockDim.x + threadIdx.x;
    if (t < n4) {
        float4 v = p[t];
        float4 bv = b[t & 15];
        p[t] = make_float4(fmaxf(v.x + bv.x, 0.f), fmaxf(v.y + bv.y, 0.f),
                           fmaxf(v.z + bv.z, 0.f), fmaxf(v.w + bv.w, 0.f));
    }
}

// mean-sum, max (uint trick; valid since h >= 0 post-ReLU), count — float4 per thread
__global__ void pool_kernel(const float* __restrict__ h, const int* __restrict__ batch,
                            float* __restrict__ gsum, unsigned* __restrict__ gmaxu,
                            float* __restrict__ gcnt, long long total /* N*16 */) {
    long long t = (long long)blockIdx.x * blockDim.x + threadIdx.x;
    if (t < total) {
        long long i = t >> 4;
        int q = (int)(t & 15);
        int g = batch[i];
        float4 v = ((const float4*)(h + i * HID))[q];
        float* gs = gsum + (long long)g * HID + q * 4;
        unsigned* gm = gmaxu + (long long)g * HID + q * 4;
        atomicAdd(gs + 0, v.x);
        atomicAdd(gs + 1, v.y);
        atomicAdd(gs + 2, v.z);
        atomicAdd(gs + 3, v.w);
        atomicMax(gm + 0, __float_as_uint(v.x));
        atomicMax(gm + 1, __float_as_uint(v.y));
        atomicMax(gm + 2, __float_as_uint(v.z));
        atomicMax(gm + 3, __float_as_uint(v.w));
        if (q == 0) atomicAdd(&gcnt[g], 1.0f);
    }
}

// fused[G,208] = [ mean(64) | max(64) | meta(64) | species_emb(16) ]
__global__ void fuse_kernel(const float* __restrict__ gsum, const float* __restrict__ gmax,
                            const float* __restrict__ gcnt, const float* __restrict__ meta,
                            const float* __restrict__ semb, const int* __restrict__ sid,
                            float* __restrict__ fused, int G) {
    int t = blockIdx.x * blockDim.x + threadIdx.x;
    if (t < G * 208) {
        int g = t / 208;
        int f = t - g * 208;
        float v;
        if (f < 64)       v = gsum[g * HID + f] / fmaxf(gcnt[g], 1.0f);
        else if (f < 128) v = gmax[g * HID + (f - 64)];
        else if (f < 192) v = meta[g * HID + (f - 128)];
        else              v = semb[sid[g] * 16 + (f - 192)];
        fused[t] = v;
    }
}

// out[g] = dot(p1[g,:64], Wp2) + bp2
__global__ void final_kernel(const float* __restrict__ p1, const float* __restrict__ Wp2,
                             const float* __restrict__ bp2, float* __restrict__ out, int G) {
    int g = blockIdx.x * blockDim.x + threadIdx.x;
    if (g < G) {
        float s = bp2[0];
#pragma unroll 8
        for (int f = 0; f < HID; ++f) s += p1[(long long)g * HID + f] * Wp2[f];
        out[g] = s;
    }
}

// ---------------- host launcher ----------------

static inline int cdiv_ll(long long a, long long b) { return (int)((a + b - 1) / b); }

extern "C" void kernel_launch(void* const* d_in, const int* in_sizes, int n_in,
                              void* d_out, int out_size, void* d_ws, size_t ws_size,
                              hipStream_t stream) {
    const float* x        = (const float*)d_in[0];
    const float* metadata = (const float*)d_in[1];
    const int*   eidx     = (const int*)d_in[2];
    const int*   batch    = (const int*)d_in[3];
    const int*   sid      = (const int*)d_in[4];
    const float* W1  = (const float*)d_in[5];
    const float* b1  = (const float*)d_in[6];
    const float* W2  = (const float*)d_in[7];
    const float* b2  = (const float*)d_in[8];
    const float* W3  = (const float*)d_in[9];
    const float* b3  = (const float*)d_in[10];
    const float* Wm  = (const float*)d_in[11];
    const float* bm  = (const float*)d_in[12];
    const float* semb = (const float*)d_in[13];
    const float* Wp1 = (const float*)d_in[14];
    const float* bp1 = (const float*)d_in[15];
    const float* Wp2 = (const float*)d_in[16];
    const float* bp2 = (const float*)d_in[17];

    const int N = in_sizes[0] / 32;     // 400000
    const int E = in_sizes[2] / 2;      // 1600000
    const int G = in_sizes[1] / 16;     // 16384
    const int* row = eidx;
    const int* col = eidx + E;

    float* ws   = (float*)d_ws;
    float* dinv = ws;  ws += N;
    float* h    = ws;  ws += (long long)N * HID;
    float* agg  = ws;  ws += (long long)N * HID;
    float* gsum = ws;  ws += (long long)G * HID;
    float* gmax = ws;  ws += (long long)G * HID;
    float* gcnt = ws;  ws += G;
    float* meta = ws;  ws += (long long)G * HID;
    float* fusd = ws;  ws += (long long)G * 208;
    float* p1   = ws;  ws += (long long)G * HID;

    const dim3 blk(256);
    const long long N16 = (long long)N * 16;   // float4 granules per feature matrix
    const long long E16 = (long long)E * 16;
    const long long GH  = (long long)G * HID;
    const long long G16 = (long long)G * 16;

    // degree -> dinv
    zero_kernel<<<1024, blk, 0, stream>>>(dinv, (long long)N);
    zero_kernel<<<1024, blk, 0, stream>>>(gsum, GH * 2 + G);   // gsum|gmax|gcnt contiguous
    deg_kernel<<<cdiv_ll(E, 256), blk, 0, stream>>>(row, dinv, E);
    dinv_kernel<<<cdiv_ll(N, 256), blk, 0, stream>>>(dinv, N);

    const int nodeBlocks = cdiv_ll((long long)N / 16, 8);      // 16-row tiles, 8 waves/block
    const int grBlocks   = cdiv_ll((long long)G / 16, 8);

    // ---- GCN layer 1 (K=32) ----
    gemm64_wmma<<<nodeBlocks, blk, 32 * HID * sizeof(float), stream>>>(x, W1, h, N, 32);
    self_bias_kernel<<<cdiv_ll(N16, 256), blk, 0, stream>>>((float4*)agg, (const float4*)h,
                                                            dinv, (const float4*)b1, N16);
    scatter_kernel<<<cdiv_ll(E16, 256), blk, 0, stream>>>(row, col, dinv, h, agg, E16);
    relu_kernel<<<cdiv_ll(N16, 256), blk, 0, stream>>>((float4*)agg, N16);

    // ---- GCN layer 2 (K=64) ----
    gemm64_wmma<<<nodeBlocks, blk, 64 * HID * sizeof(float), stream>>>(agg, W2, h, N, 64);
    self_bias_kernel<<<cdiv_ll(N16, 256), blk, 0, stream>>>((float4*)agg, (const float4*)h,
                                                            dinv, (const float4*)b2, N16);
    scatter_kernel<<<cdiv_ll(E16, 256), blk, 0, stream>>>(row, col, dinv, h, agg, E16);
    relu_kernel<<<cdiv_ll(N16, 256), blk, 0, stream>>>((float4*)agg, N16);

    // ---- GCN layer 3 (K=64) ----
    gemm64_wmma<<<nodeBlocks, blk, 64 * HID * sizeof(float), stream>>>(agg, W3, h, N, 64);
    self_bias_kernel<<<cdiv_ll(N16, 256), blk, 0, stream>>>((float4*)agg, (const float4*)h,
                                                            dinv, (const float4*)b3, N16);
    scatter_kernel<<<cdiv_ll(E16, 256), blk, 0, stream>>>(row, col, dinv, h, agg, E16);
    relu_kernel<<<cdiv_ll(N16, 256), blk, 0, stream>>>((float4*)agg, N16);

    // ---- pooling ----
    pool_kernel<<<cdiv_ll(N16, 256), blk, 0, stream>>>(agg, batch, gsum, (unsigned*)gmax,
                                                       gcnt, N16);

    // ---- meta encoder (K=16) ----
    gemm64_wmma<<<grBlocks, blk, 16 * HID * sizeof(float), stream>>>(metadata, Wm, meta, G, 16);
    bias_relu_kernel<<<cdiv_ll(G16, 256), blk, 0, stream>>>((float4*)meta, (const float4*)bm,
                                                            G16);

    // ---- fusion + predictor (K=208) ----
    fuse_kernel<<<cdiv_ll((long long)G * 208, 256), blk, 0, stream>>>(gsum, gmax, gcnt, meta,
                                                                      semb, sid, fusd, G);
    gemm64_wmma<<<grBlocks, blk, 208 * HID * sizeof(float), stream>>>(fusd, Wp1, p1, G, 208);
    bias_relu_kernel<<<cdiv_ll(G16, 256), blk, 0, stream>>>((float4*)p1, (const float4*)bp1,
                                                            G16);
    final_kernel<<<cdiv_ll(G, 256), blk, 0, stream>>>(p1, Wp2, bp2, (float*)d_out, G);
}